// DifferentiableConstructor_80590766342365
// MI455X (gfx1250) — compile-verified
//
#include <hip/hip_runtime.h>
#include <math.h>

// ---------------- problem constants ----------------
static constexpr int S   = 512;
static constexpr int Dm  = 128;
static constexpr int Hh  = 4;
static constexpr int DHd = 32;
static constexpr int DFh = 512;
static constexpr int DNn = 64;
static constexpr int DEe = 64;
static constexpr int SM1 = 511;            // S - 1
static constexpr int EE  = S * SM1;        // 261632 edges

typedef float v2f __attribute__((ext_vector_type(2)));
typedef float v8f __attribute__((ext_vector_type(8)));

__device__ __forceinline__ float gelu_exact(float v) {
  return 0.5f * v * (1.0f + erff(v * 0.7071067811865476f));
}

// ---------------- backbone kernels (tiny FLOPs, exact fp32) ----------------

__global__ void embed_pe_kernel(const int* __restrict__ tok,
                                const float* __restrict__ emb,
                                float* __restrict__ x) {
  int idx = blockIdx.x * blockDim.x + threadIdx.x;   // S*Dm
  if (idx >= S * Dm) return;
  int s = idx / Dm, d = idx - s * Dm;
  int j = d >> 1;
  float div = expf(-(float)(2 * j) * (9.210340371976184f / 128.0f));
  float ang = (float)s * div;
  float pe  = (d & 1) ? cosf(ang) : sinf(ang);
  x[idx] = emb[tok[s] * Dm + d] + pe;
}

__global__ void gemm_bias_kernel(const float* __restrict__ A,
                                 const float* __restrict__ W,
                                 const float* __restrict__ b,
                                 float* __restrict__ C,
                                 int M, int K, int N, int act) {
  int idx = blockIdx.x * blockDim.x + threadIdx.x;
  if (idx >= M * N) return;
  int m = idx / N, n = idx - m * N;
  const float* a = A + (size_t)m * K;
  float acc = b[n];
  for (int k = 0; k < K; ++k) acc = fmaf(a[k], W[(size_t)k * N + n], acc);
  if (act == 1) acc = gelu_exact(acc);
  C[idx] = acc;
}

__global__ void attn_scores_kernel(const float* __restrict__ qkv,
                                   float* __restrict__ scores) {
  int idx = blockIdx.x * blockDim.x + threadIdx.x;   // Hh*S*S
  if (idx >= Hh * S * S) return;
  int h   = idx / (S * S);
  int rem = idx - h * (S * S);
  int q = rem / S, k = rem - q * S;
  const float* qp = qkv + (size_t)q * (3 * Dm) + h * DHd;
  const float* kp = qkv + (size_t)k * (3 * Dm) + Dm + h * DHd;
  float acc = 0.f;
  for (int d = 0; d < DHd; ++d) acc = fmaf(qp[d], kp[d], acc);
  scores[idx] = acc * 0.17677669529663687f;          // 1/sqrt(32)
}

__global__ void softmax_kernel(float* __restrict__ scores) {
  int row = blockIdx.x * blockDim.x + threadIdx.x;   // Hh*S rows
  if (row >= Hh * S) return;
  float* p = scores + (size_t)row * S;
  float mx = -3.4e38f;
  for (int i = 0; i < S; ++i) mx = fmaxf(mx, p[i]);
  float sum = 0.f;
  for (int i = 0; i < S; ++i) { float e = expf(p[i] - mx); p[i] = e; sum += e; }
  float inv = 1.0f / sum;
  for (int i = 0; i < S; ++i) p[i] *= inv;
}

__global__ void attn_out_kernel(const float* __restrict__ scores,
                                const float* __restrict__ qkv,
                                float* __restrict__ o) {
  int idx = blockIdx.x * blockDim.x + threadIdx.x;   // S*Dm
  if (idx >= S * Dm) return;
  int q = idx / Dm, dm = idx - q * Dm;
  int h = dm / DHd;
  const float* sp = scores + ((size_t)h * S + q) * S;
  float acc = 0.f;
  for (int k = 0; k < S; ++k)
    acc = fmaf(sp[k], qkv[(size_t)k * (3 * Dm) + 2 * Dm + dm], acc);
  o[idx] = acc;
}

__global__ void add_ln_kernel(float* __restrict__ x, const float* __restrict__ t,
                              const float* __restrict__ g, const float* __restrict__ b) {
  int row = blockIdx.x * blockDim.x + threadIdx.x;
  if (row >= S) return;
  float* xr = x + (size_t)row * Dm;
  const float* tr = t + (size_t)row * Dm;
  float mean = 0.f;
  for (int i = 0; i < Dm; ++i) { float s = xr[i] + tr[i]; xr[i] = s; mean += s; }
  mean *= (1.0f / Dm);
  float var = 0.f;
  for (int i = 0; i < Dm; ++i) { float d0 = xr[i] - mean; var = fmaf(d0, d0, var); }
  var *= (1.0f / Dm);
  float inv = rsqrtf(var + 1e-5f);
  for (int i = 0; i < Dm; ++i) xr[i] = (xr[i] - mean) * inv * g[i] + b[i];
}

__global__ void edge_index_kernel(float* __restrict__ out) {
  int e = blockIdx.x * blockDim.x + threadIdx.x;
  if (e >= EE) return;
  int s  = e / SM1;
  int r  = e - s * SM1;
  int tg = r + (r >= s ? 1 : 0);
  out[e]       = (float)s;
  out[EE + e]  = (float)tg;
}

// ---------------- edge MLP: WMMA f32 16x16x4, two M-tiles per wave ----------------
// Each wave owns 32 edges (two 16-row M-tiles) so every LDS-loaded B fragment feeds
// two WMMAs. K = 257 = 4 LDS-staged K-panels of 64 (WMMA) + pos_diff rank-1 (VALU).
// Panels 0-1 read x[src], panels 2-3 read x[tgt] (the K=128 split is panel-aligned),
// so the A-source select is hoisted out of the inner loop (branchless steady state).

__device__ __forceinline__ void edge_epilogue(
    const v8f* __restrict__ acc1, const v8f* __restrict__ acc2,
    int gbase, int half, int mlo,
    const float* __restrict__ w1, const float* __restrict__ b1s,
    const float* __restrict__ w2s, float b2v,
    const float* __restrict__ te, const float* __restrict__ tebs,
    const float* __restrict__ u,
    float* __restrict__ out_feats, float* __restrict__ out_gates) {
  // C-fragment layout: lane L, reg r -> M = r + 8*half, N = nb*16 + mlo
  float pdv[8];
#pragma unroll
  for (int r = 0; r < 8; ++r) {
    int e_r = gbase + half * 8 + r;
    int sr  = e_r / SM1;
    int rr  = e_r - sr * SM1;
    int tr  = rr + (rr >= sr ? 1 : 0);
    pdv[r]  = (float)(sr - tr) * (1.0f / (float)S);   // pos_diff (K=256 column)
  }

  float part[8];
#pragma unroll
  for (int r = 0; r < 8; ++r) part[r] = 0.f;

#pragma unroll
  for (int nb = 0; nb < 8; ++nb) {
    float w1l = w1[256 * 128 + nb * 16 + mlo];        // K=256 row of sc_W1
    float bia = b1s[nb * 16 + mlo];
    float w2v = w2s[nb * 16 + mlo];
#pragma unroll
    for (int r = 0; r < 8; ++r) {
      float hv = acc1[nb][r] + pdv[r] * w1l + bia;
      part[r]  = fmaf(gelu_exact(hv), w2v, part[r]);  // partial logit over N subset
    }
  }

  float gate[8];
#pragma unroll
  for (int r = 0; r < 8; ++r) {
    float v = part[r];                                // 16-lane reduce (stays in half)
    v += __shfl_xor(v, 1, 32);
    v += __shfl_xor(v, 2, 32);
    v += __shfl_xor(v, 4, 32);
    v += __shfl_xor(v, 8, 32);
    int e_r  = gbase + half * 8 + r;
    float uu = u[e_r];
    uu = fminf(fmaxf(uu, 1e-8f), 1.0f - 1e-8f);
    float gn = -logf(-logf(uu));                      // gumbel noise
    float z  = v + b2v + gn;                          // TEMP = 1.0
    gate[r]  = 1.0f / (1.0f + expf(-z));
  }

  if (mlo == 0) {
#pragma unroll
    for (int r = 0; r < 8; ++r) out_gates[gbase + half * 8 + r] = gate[r];
  }

#pragma unroll
  for (int nb = 0; nb < 4; ++nb) {
    float tel = te[256 * 64 + nb * 16 + mlo];         // K=256 row of te_W
    float bia = tebs[nb * 16 + mlo];
#pragma unroll
    for (int r = 0; r < 8; ++r) {
      float ef = (acc2[nb][r] + pdv[r] * tel + bia) * gate[r];
      out_feats[(size_t)(gbase + half * 8 + r) * DEe + nb * 16 + mlo] = ef;
    }
  }
}

__global__ __launch_bounds__(128) void edge_mlp_kernel(
    const float* __restrict__ x,
    const float* __restrict__ w1, const float* __restrict__ b1,
    const float* __restrict__ w2, const float* __restrict__ b2,
    const float* __restrict__ te, const float* __restrict__ teb,
    const float* __restrict__ u,
    float* __restrict__ out_feats, float* __restrict__ out_gates) {
  __shared__ float w1s[64 * 128];
  __shared__ float tes[64 * 64];
  __shared__ float w2s[128];
  __shared__ float b1s[128];
  __shared__ float tebs[64];

  const int t    = threadIdx.x;
  const int lane = t & 31;
  const int wave = t >> 5;
  const int half = lane >> 4;      // 0 or 1
  const int mlo  = lane & 15;
  const int base = blockIdx.x * 128 + wave * 32;   // 32 edges per wave

  if (t < 128) { w2s[t] = w2[t]; b1s[t] = b1[t]; }
  if (t < 64)  { tebs[t] = teb[t]; }

  // A-fragment rows for this lane: edges base+mlo (tile 0) and base+16+mlo (tile 1)
  int e0 = base + mlo;
  int s0 = e0 / SM1; int r0 = e0 - s0 * SM1; int t0 = r0 + (r0 >= s0 ? 1 : 0);
  int e1 = e0 + 16;
  int s1 = e1 / SM1; int r1 = e1 - s1 * SM1; int t1 = r1 + (r1 >= s1 ? 1 : 0);
  const float* xs0 = x + (size_t)s0 * Dm;
  const float* xt0 = x + (size_t)t0 * Dm;
  const float* xs1 = x + (size_t)s1 * Dm;
  const float* xt1 = x + (size_t)t1 * Dm;

  v8f vz = {0.f, 0.f, 0.f, 0.f, 0.f, 0.f, 0.f, 0.f};
  v8f acc1a[8], acc1b[8], acc2a[4], acc2b[4];
#pragma unroll
  for (int i = 0; i < 8; ++i) { acc1a[i] = vz; acc1b[i] = vz; }
#pragma unroll
  for (int i = 0; i < 4; ++i) { acc2a[i] = vz; acc2b[i] = vz; }

  for (int kp = 0; kp < 4; ++kp) {
    __syncthreads();
    // cooperative panel load (coalesced b32s, L2-resident weights)
    for (int i = 0; i < 64; ++i)
      w1s[i * 128 + t] = w1[(size_t)(kp * 64 + i) * 128 + t];
    for (int idx = t; idx < 64 * 64; idx += 128)
      tes[idx] = te[(size_t)(kp * 64 + (idx >> 6)) * 64 + (idx & 63)];
    __syncthreads();

    // prefetch next panel into cache while this one computes (global_prefetch_b8)
    if (kp < 3) {
      __builtin_prefetch(&w1[(size_t)((kp + 1) * 64 + (t >> 1)) * 128 + (t & 1) * 64], 0, 1);
      __builtin_prefetch(&te[(size_t)((kp + 1) * 64 + (t >> 1)) * 64 + (t & 1) * 32], 0, 1);
    }

    // Panel-aligned src/tgt split: hoist A-pointer select out of the inner loop.
    const float* pa0 = (kp < 2) ? (xs0 + kp * 64) : (xt0 + (kp - 2) * 64);
    const float* pa1 = (kp < 2) ? (xs1 + kp * 64) : (xt1 + (kp - 2) * 64);

    for (int kc = 0; kc < 16; ++kc) {
      // A frags: lane holds p[m][kg], p[m][kg+1];  kg = chunk_k0 + 2*(lane/16)
      int ko = kc * 4 + half * 2;                    // 8B-aligned by construction
      v2f a0 = *(const v2f*)(pa0 + ko);
      v2f a1 = *(const v2f*)(pa1 + ko);
      int kr = ko;                                   // same offset inside LDS panel
#pragma unroll
      for (int nb = 0; nb < 8; ++nb) {        // scorer N-tiles (128 cols)
        v2f bf;
        bf.x = w1s[kr * 128 + nb * 16 + mlo];
        bf.y = w1s[(kr + 1) * 128 + nb * 16 + mlo];
        acc1a[nb] = __builtin_amdgcn_wmma_f32_16x16x4_f32(
            false, a0, false, bf, (short)0, acc1a[nb], false, false);
        acc1b[nb] = __builtin_amdgcn_wmma_f32_16x16x4_f32(
            false, a1, false, bf, (short)0, acc1b[nb], false, false);
      }
#pragma unroll
      for (int nb = 0; nb < 4; ++nb) {        // te_W N-tiles (64 cols)
        v2f bf;
        bf.x = tes[kr * 64 + nb * 16 + mlo];
        bf.y = tes[(kr + 1) * 64 + nb * 16 + mlo];
        acc2a[nb] = __builtin_amdgcn_wmma_f32_16x16x4_f32(
            false, a0, false, bf, (short)0, acc2a[nb], false, false);
        acc2b[nb] = __builtin_amdgcn_wmma_f32_16x16x4_f32(
            false, a1, false, bf, (short)0, acc2b[nb], false, false);
      }
    }
  }

  float b2v = b2[0];
  edge_epilogue(acc1a, acc2a, base,      half, mlo, w1, b1s, w2s, b2v, te, tebs, u,
                out_feats, out_gates);
  edge_epilogue(acc1b, acc2b, base + 16, half, mlo, w1, b1s, w2s, b2v, te, tebs, u,
                out_feats, out_gates);
}

// ---------------- host launcher ----------------

extern "C" void kernel_launch(void* const* d_in, const int* in_sizes, int n_in,
                              void* d_out, int out_size, void* d_ws, size_t ws_size,
                              hipStream_t stream) {
  const int*   tok   = (const int*)d_in[0];
  const float* u     = (const float*)d_in[1];
  const float* emb   = (const float*)d_in[2];
  const float* Wqkv  = (const float*)d_in[3];
  const float* bqkv  = (const float*)d_in[4];
  const float* Wo    = (const float*)d_in[5];
  const float* bo    = (const float*)d_in[6];
  const float* ln1g  = (const float*)d_in[7];
  const float* ln1b  = (const float*)d_in[8];
  const float* Wff1  = (const float*)d_in[9];
  const float* bff1  = (const float*)d_in[10];
  const float* Wff2  = (const float*)d_in[11];
  const float* bff2  = (const float*)d_in[12];
  const float* ln2g  = (const float*)d_in[13];
  const float* ln2b  = (const float*)d_in[14];
  const float* nodeW = (const float*)d_in[15];
  const float* nodeb = (const float*)d_in[16];
  const float* scW1  = (const float*)d_in[17];
  const float* scb1  = (const float*)d_in[18];
  const float* scW2  = (const float*)d_in[19];
  const float* scb2  = (const float*)d_in[20];
  const float* teW   = (const float*)d_in[21];
  const float* teB   = (const float*)d_in[22];

  float* ws     = (float*)d_ws;
  float* x      = ws;                        // S*Dm
  float* qkv    = x + S * Dm;                // S*3Dm
  float* scores = qkv + S * 3 * Dm;          // Hh*S*S
  float* o      = scores + Hh * S * S;       // S*Dm
  float* t1     = o + S * Dm;                // S*DFh
  float* t2     = t1 + S * DFh;              // S*Dm

  float* outp      = (float*)d_out;
  float* node_out  = outp;                                   // S*DNn
  float* feats_out = outp + S * DNn;                         // EE*DEe
  float* eidx_out  = feats_out + (size_t)EE * DEe;           // 2*EE
  float* gates_out = eidx_out + 2 * (size_t)EE;              // EE

  embed_pe_kernel<<<(S * Dm + 255) / 256, 256, 0, stream>>>(tok, emb, x);
  for (int l = 0; l < 2; ++l) {
    gemm_bias_kernel<<<(S * 3 * Dm + 255) / 256, 256, 0, stream>>>(
        x, Wqkv + (size_t)l * Dm * 3 * Dm, bqkv + l * 3 * Dm, qkv, S, Dm, 3 * Dm, 0);
    attn_scores_kernel<<<(Hh * S * S + 255) / 256, 256, 0, stream>>>(qkv, scores);
    softmax_kernel<<<(Hh * S + 255) / 256, 256, 0, stream>>>(scores);
    attn_out_kernel<<<(S * Dm + 255) / 256, 256, 0, stream>>>(scores, qkv, o);
    gemm_bias_kernel<<<(S * Dm + 255) / 256, 256, 0, stream>>>(
        o, Wo + (size_t)l * Dm * Dm, bo + l * Dm, t2, S, Dm, Dm, 0);
    add_ln_kernel<<<(S + 255) / 256, 256, 0, stream>>>(x, t2, ln1g + l * Dm, ln1b + l * Dm);
    gemm_bias_kernel<<<(S * DFh + 255) / 256, 256, 0, stream>>>(
        x, Wff1 + (size_t)l * Dm * DFh, bff1 + l * DFh, t1, S, Dm, DFh, 1);
    gemm_bias_kernel<<<(S * Dm + 255) / 256, 256, 0, stream>>>(
        t1, Wff2 + (size_t)l * DFh * Dm, bff2 + l * Dm, t2, S, DFh, Dm, 0);
    add_ln_kernel<<<(S + 255) / 256, 256, 0, stream>>>(x, t2, ln2g + l * Dm, ln2b + l * Dm);
  }
  gemm_bias_kernel<<<(S * DNn + 255) / 256, 256, 0, stream>>>(x, nodeW, nodeb, node_out, S, Dm, DNn, 0);
  edge_index_kernel<<<(EE + 255) / 256, 256, 0, stream>>>(eidx_out);
  edge_mlp_kernel<<<EE / 128, 128, 0, stream>>>(
      x, scW1, scb1, scW2, scb2, teW, teB, u, feats_out, gates_out);
}